// Mixer_79431125172594
// MI455X (gfx1250) — compile-verified
//
#include <hip/hip_runtime.h>
#include <math.h>

typedef __attribute__((ext_vector_type(2))) float v2f;
typedef __attribute__((ext_vector_type(8))) float v8f;

#define SRATE 44100.0f
#define NT    65536
#define NCH   32

// Per-lane delay-line table: lanes 0..7 = combs, 8..11 = allpasses,
// 12..31 = private dummy slots (keeps load/store branch- and conflict-free).
__device__ __constant__ int kLen32[32] = {
    1116,1188,1277,1356,1422,1491,1557,1617,   // combs
     556, 441, 341, 225,                       // allpasses
       1,1,1,1,1,1,1,1,1,1,1,1,1,1,1,1,1,1,1,1 // dummies
};
__device__ __constant__ int kOff32[32] = {
        0, 1116, 2304, 3581, 4937, 6359, 7850, 9407,   // combs (sum 11024)
    11024,11580,12021,12362,                           // allpasses (sum 1563)
    12587,12588,12589,12590,12591,12592,12593,12594,   // dummies
    12595,12596,12597,12598,12599,12600,12601,12602,
    12603,12604,12605,12606
};
#define BUF_FLOATS 12607

// ---------------------------------------------------------------------------
// Kernel 1: per-channel sequential DSP chain. One wave32 per channel.
// One ds_load + one ds_store per sample services all 12 delay lines.
// ---------------------------------------------------------------------------
__global__ __launch_bounds__(32)
void dsp_kernel(const float* __restrict__ x, const float* __restrict__ p,
                float* __restrict__ xf, float* __restrict__ pout)
{
    __shared__ float buf[BUF_FLOATS];

    const int ch   = blockIdx.x;   // 0..31  (b = ch/8, n = ch%8)
    const int lane = threadIdx.x;  // 0..31

    for (int i = lane; i < BUF_FLOATS; i += 32) buf[i] = 0.f;
    __syncthreads();

    const float* pf = p + ch * 24;
    float pv[24];
#pragma unroll
    for (int i = 0; i < 24; ++i) pv[i] = pf[i];

    // restore() of all parameters
    const float gain_in  = pv[0]*60.f  - 48.f;
    const float hp_fc    = pv[1]*350.f;
    const float lp_fc    = pv[2]*19000.f + 3000.f;
    const float hs_fc    = pv[3]*14500.f + 1500.f;
    const float hs_g     = pv[4]*30.f  - 15.f;
    const float ls_fc    = pv[5]*420.f + 30.f;
    const float ls_g     = pv[6]*30.f  - 15.f;
    const float mh_fc    = pv[7]*6400.f + 600.f;
    const float mh_g     = pv[8]*30.f  - 15.f;
    const float mh_q     = pv[9]*2.5f  + 0.5f;
    const float ml_fc    = pv[10]*2300.f + 200.f;
    const float ml_g     = pv[11]*30.f - 15.f;
    const float ml_q     = pv[12]*2.5f + 0.5f;
    const float c_thr    = pv[13]*30.f - 20.f;
    const float c_rat    = pv[14]*19.f + 1.f;
    const float c_att    = pv[15]*29.f + 1.f;
    const float c_rel    = pv[16]*3900.f + 100.f;
    const float gain_out = pv[22]*60.f - 48.f;
    const float pan      = pv[23]*0.4f + 0.3f;

    if (lane == 0) {
        float po[24] = {gain_in,hp_fc,lp_fc,hs_fc,hs_g,ls_fc,ls_g,mh_fc,mh_g,mh_q,
                        ml_fc,ml_g,ml_q,c_thr,c_rat,c_att,c_rel,
                        pv[17],pv[18],pv[19],pv[20],pv[21],gain_out,pan};
        for (int i = 0; i < 24; ++i) pout[ch*24 + i] = po[i];
    }

    // ---- biquad cascade coefficients (a0-normalized); setup is one-shot,
    //      full-accuracy libm is fine here ----
    float B0[6], B1[6], B2[6], A1c[6], A2c[6];
    {
        const float PIf = 3.14159265358979f;
        float K = tanf(PIf * hp_fc / SRATE);                 // sec0: 1st-order HP
        float b0 = 1.f / (1.f + K);
        B0[0]=b0; B1[0]=-b0; B2[0]=0.f; A1c[0]=(K-1.f)/(K+1.f); A2c[0]=0.f;

        K = tanf(PIf * lp_fc / SRATE);                       // sec1: 1st-order LP
        b0 = K / (1.f + K);
        B0[1]=b0; B1[1]=b0; B2[1]=0.f; A1c[1]=(K-1.f)/(K+1.f); A2c[1]=0.f;

        {   // sec2: high shelf (s=+1)
            float A = exp10f(hs_g*0.025f);
            float w = 2.f*PIf*hs_fc/SRATE, cw = cosf(w);
            float alpha = sinf(w)*0.7071067811865476f;
            float sq = 2.f*sqrtf(A)*alpha;
            float b0s = A*((A+1.f)+(A-1.f)*cw+sq);
            float b1s = -2.f*A*((A-1.f)+(A+1.f)*cw);
            float b2s = A*((A+1.f)+(A-1.f)*cw-sq);
            float a0s = (A+1.f)-(A-1.f)*cw+sq;
            float a1s = 2.f*((A-1.f)-(A+1.f)*cw);
            float a2s = (A+1.f)-(A-1.f)*cw-sq;
            float inv = 1.f/a0s;
            B0[2]=b0s*inv; B1[2]=b1s*inv; B2[2]=b2s*inv; A1c[2]=a1s*inv; A2c[2]=a2s*inv;
        }
        {   // sec3: low shelf (s=-1)
            float A = exp10f(ls_g*0.025f);
            float w = 2.f*PIf*ls_fc/SRATE, cw = cosf(w);
            float alpha = sinf(w)*0.7071067811865476f;
            float sq = 2.f*sqrtf(A)*alpha;
            float b0s = A*((A+1.f)-(A-1.f)*cw+sq);
            float b1s = 2.f*A*((A-1.f)-(A+1.f)*cw);
            float b2s = A*((A+1.f)-(A-1.f)*cw-sq);
            float a0s = (A+1.f)+(A-1.f)*cw+sq;
            float a1s = -2.f*((A-1.f)+(A+1.f)*cw);
            float a2s = (A+1.f)+(A-1.f)*cw-sq;
            float inv = 1.f/a0s;
            B0[3]=b0s*inv; B1[3]=b1s*inv; B2[3]=b2s*inv; A1c[3]=a1s*inv; A2c[3]=a2s*inv;
        }
        {   // sec4: peak (mid-high)
            float A = exp10f(mh_g*0.025f);
            float w = 2.f*PIf*mh_fc/SRATE, cw = cosf(w);
            float alpha = sinf(w)/(2.f*mh_q);
            float a0s = 1.f + alpha/A, inv = 1.f/a0s;
            B0[4]=(1.f+alpha*A)*inv; B1[4]=(-2.f*cw)*inv; B2[4]=(1.f-alpha*A)*inv;
            A1c[4]=(-2.f*cw)*inv;    A2c[4]=(1.f-alpha/A)*inv;
        }
        {   // sec5: peak (mid-low)
            float A = exp10f(ml_g*0.025f);
            float w = 2.f*PIf*ml_fc/SRATE, cw = cosf(w);
            float alpha = sinf(w)/(2.f*ml_q);
            float a0s = 1.f + alpha/A, inv = 1.f/a0s;
            B0[5]=(1.f+alpha*A)*inv; B1[5]=(-2.f*cw)*inv; B2[5]=(1.f-alpha*A)*inv;
            A1c[5]=(-2.f*cw)*inv;    A2c[5]=(1.f-alpha/A)*inv;
        }
    }

    // derived runtime constants
    const float gin    = exp10f(gain_in  * 0.05f);
    const float gout   = exp10f(gain_out * 0.05f);
    const float a_att  = expf(-1.f / (c_att * 0.001f * SRATE));
    const float a_rel  = expf(-1.f / (c_rel * 0.001f * SRATE));
    const float ratInv = 1.f - 1.f / c_rat;
    const float fbk    = pv[17]*0.28f + 0.7f;
    const float damp   = pv[18]*0.4f;
    const float omd    = 1.f - damp;
    const float wet1   = 1.5f * pv[19] * (1.f + pv[21]);
    const float dryg   = 2.f * pv[20];

    // per-lane delay-line slot
    const int blen = kLen32[lane];
    const int boff = kOff32[lane];
    int bidx = 0;
    float cst = 0.f;                 // comb damping state (lanes 0..7)

    float z1[6] = {0,0,0,0,0,0}, z2[6] = {0,0,0,0,0,0};
    float g = 0.f;                   // compressor gain state (dB)

    const float* xin = x  + ch * NT;
    float*       xo  = xf + ch * NT;
    float xv = 0.f, ov = 0.f;        // staged, coalesced I/O

#pragma unroll 2
    for (int t = 0; t < NT; ++t) {
        const int sub = t & 31;
        if (sub == 0) xv = xin[t + lane];

        // delay-line read for this sample (independent of x -> overlaps the
        // filter/compressor chain): lanes 0..7 comb, 8..11 allpass, rest dummy
        const float bval = buf[boff + bidx];

        float s = __shfl(xv, sub) * gin;

        // 6 cascaded biquads (transposed DF-II). Critical path = 6 FMAs
        // (z-updates have a full iteration of slack).
#pragma unroll
        for (int k = 0; k < 6; ++k) {
            float y = fmaf(B0[k], s, z1[k]);
            z1[k] = fmaf(B1[k], s, fmaf(-A1c[k], y, z2[k]));
            z2[k] = fmaf(B2[k], s, -A2c[k] * y);
            s = y;
        }

        // compressor (fast HW transcendentals: v_log_f32 / v_exp_f32)
        float lvl  = 8.6858896381f * __logf(fabsf(s) + 1e-6f); // 20*log10
        float tgt  = fminf(0.f, (c_thr - lvl) * ratInv);
        float coef = (tgt < g) ? a_att : a_rel;
        g = fmaf(coef, g, (1.f - coef) * tgt);
        s = s * __expf(g * 0.11512925465f);                    // 10^(g/20)

        // freeverb combs: lanes 0..7 own one comb each
        const float inp = s * 0.015f;
        cst = fmaf(bval, omd, cst * damp);
        const float wcomb = fmaf(cst, fbk, inp);

        // sum of the 8 comb outputs -> broadcast
        float red = (lane < 8) ? bval : 0.f;
        red += __shfl_xor(red, 1);
        red += __shfl_xor(red, 2);
        red += __shfl_xor(red, 4);
        float a0 = __shfl(red, 0);

        // 4 allpasses: reads already done in parallel (lanes 8..11);
        // the serial part is pure VALU algebra.
        const float b0 = __shfl(bval,  8);
        const float b1 = __shfl(bval,  9);
        const float b2 = __shfl(bval, 10);
        const float b3 = __shfl(bval, 11);
        float w0 = fmaf(b0, 0.5f, a0); float a1 = b0 - a0;
        float w1 = fmaf(b1, 0.5f, a1); float a2 = b1 - a1;
        float w2 = fmaf(b2, 0.5f, a2); float a3 = b2 - a2;
        float w3 = fmaf(b3, 0.5f, a3); float a4 = b3 - a3;

        // single store services combs + allpasses (dummies for lanes >=12)
        float apw  = (lane == 8) ? w0 : (lane == 9) ? w1 : (lane == 10) ? w2 : w3;
        float sval = (lane < 8) ? wcomb : apw;
        buf[boff + bidx] = sval;
        bidx = (bidx + 1 == blen) ? 0 : bidx + 1;

        float val = fmaf(a4, wet1, s * dryg) * gout;
        if (sub == lane) ov = val;                    // stage into my slot
        if (sub == 31)   xo[(t & ~31) + lane] = ov;   // coalesced 128B store
    }
}

// ---------------------------------------------------------------------------
// Kernel 2: stereo pan-mix, y(2xT) = P(2x8) * X(8xT) per batch, via
// V_WMMA_F32_16X16X4_F32.  One wave handles 8 consecutive 16-sample tiles of
// one batch; the A-matrix (pan gains) is built once per wave.
// ---------------------------------------------------------------------------
__global__ __launch_bounds__(32)
void pan_mix_kernel(const float* __restrict__ xf, const float* __restrict__ p,
                    float* __restrict__ out)
{
    const int wave  = blockIdx.x;        // 0..2047
    const int b     = wave >> 9;         // 512 waves per batch
    const int tbase = (wave & 511) * 128;
    const int lane  = threadIdx.x;
    const int h     = lane >> 4;         // half-wave
    const int l     = lane & 15;

    // A[m=l][k=2h+j] = pan gain of track (4g+k) for channel m (rows m>=2 zero)
    v2f A[2];
#pragma unroll
    for (int grp = 0; grp < 2; ++grp) {
#pragma unroll
        for (int j = 0; j < 2; ++j) {
            const int k     = 2*h + j;
            const int track = 4*grp + k;
            float pan = p[(b*8 + track)*24 + 23] * 0.4f + 0.3f;
            float th  = pan * 1.5707963267948966f;
            A[grp][j] = (l == 0) ? __cosf(th) : ((l == 1) ? __sinf(th) : 0.0f);
        }
    }

#pragma unroll
    for (int tile = 0; tile < 8; ++tile) {
        const int t0 = tbase + tile * 16;
        v8f acc = {};
#pragma unroll
        for (int grp = 0; grp < 2; ++grp) {
            v2f bm;
#pragma unroll
            for (int j = 0; j < 2; ++j) {
                const int track = 4*grp + 2*h + j;     // B[k][n=l]
                bm[j] = xf[(b*8 + track)*NT + t0 + l];
            }
            acc = __builtin_amdgcn_wmma_f32_16x16x4_f32(
                      false, A[grp], false, bm, (short)0, acc, false, false);
        }
        // D rows: m = vgpr + 8h; valid outputs m=0,1 live in half h==0
        if (h == 0) {
            out[(b*2 + 0)*NT + t0 + l] = acc[0];
            out[(b*2 + 1)*NT + t0 + l] = acc[1];
        }
    }
}

// ---------------------------------------------------------------------------
extern "C" void kernel_launch(void* const* d_in, const int* in_sizes, int n_in,
                              void* d_out, int out_size, void* d_ws, size_t ws_size,
                              hipStream_t stream)
{
    const float* x = (const float*)d_in[0];   // (4,8,65536)
    const float* p = (const float*)d_in[1];   // (4,8,24)
    float* out = (float*)d_out;               // y (4,2,65536) then p_out (4,8,24)
    float* xf  = (float*)d_ws;                // 32*65536 floats = 8 MB scratch

    float* pout = out + 4*2*NT;               // offset 524288

    dsp_kernel<<<NCH, 32, 0, stream>>>(x, p, xf, pout);
    pan_mix_kernel<<<2048, 32, 0, stream>>>(xf, p, out);
}